// Gate_5523327943229
// MI455X (gfx1250) — compile-verified
//
#include <hip/hip_runtime.h>
#include <hip/hip_bf16.h>

// MoE router gate for MI455X (gfx1250, wave32).
// GEMM: V_WMMA_F32_16X16X4_F32, one wave = 16 tokens x 64 experts.
// x read once from HBM (512 MB -> ~22us floor @ 23.3 TB/s); weight cache-hot.

#define DIMK        2048
#define NE          64
#define NTOPK       6
#define NGRP        8
#define EPG         8      // experts per group
#define TG          4      // groups kept
#define NTOKENS     65536
#define ROUTE_SCALE_F 1.0f
#define BAL_F       1e-5f

typedef __attribute__((ext_vector_type(2))) float v2f;
typedef __attribute__((ext_vector_type(8))) float v8f;

__global__ void zero_ws_kernel(float* __restrict__ p) {
    p[threadIdx.x] = 0.0f;   // 128 floats: counts[64] + psum[64]
}

__global__ __launch_bounds__(256)
void moe_gate_wmma_kernel(const float* __restrict__ x,
                          const float* __restrict__ weight,
                          const float* __restrict__ bias,
                          float* __restrict__ outW,
                          int*   __restrict__ outI,
                          float* __restrict__ gCnt,
                          float* __restrict__ gP)
{
    __shared__ float sLogits[8][16][NE];   // 32 KB: per-wave 16 tokens x 64 experts
    __shared__ float sCnt[NE];
    __shared__ float sP[NE];
    __shared__ float sBias[NE];

    const int tid  = threadIdx.x;
    const int wv   = tid >> 5;     // wave id 0..7
    const int lane = tid & 31;
    const int row  = lane & 15;    // token-in-tile (A) / expert-in-slice (B)
    const int hi   = lane >> 4;    // half-wave selects K-pair {0,1} vs {2,3}

    if (tid < NE) { sCnt[tid] = 0.0f; sP[tid] = 0.0f; sBias[tid] = bias[tid]; }
    __syncthreads();

    const int tile = blockIdx.x * 8 + wv;  // 16-token tile index
    const int t0   = tile * 16;

    // A fragment: lanes 0-15 hold (K, K+1), lanes 16-31 hold (K+2, K+3) -> +2*hi bias
    const float* xp  = x + (size_t)(t0 + row) * DIMK + 2 * hi;
    // B fragments (weight.T): lane n holds column (expert) n of the slice
    const float* w0p = weight + (size_t)(row +  0) * DIMK + 2 * hi;
    const float* w1p = weight + (size_t)(row + 16) * DIMK + 2 * hi;
    const float* w2p = weight + (size_t)(row + 32) * DIMK + 2 * hi;
    const float* w3p = weight + (size_t)(row + 48) * DIMK + 2 * hi;

    v8f c0 = {}, c1 = {}, c2 = {}, c3 = {};
#pragma unroll 4
    for (int k = 0; k < DIMK; k += 4) {
        v2f a  = *(const v2f*)(xp  + k);
        v2f b0 = *(const v2f*)(w0p + k);
        v2f b1 = *(const v2f*)(w1p + k);
        v2f b2 = *(const v2f*)(w2p + k);
        v2f b3 = *(const v2f*)(w3p + k);
        c0 = __builtin_amdgcn_wmma_f32_16x16x4_f32(false, a, false, b0, (short)0, c0, false, false);
        c1 = __builtin_amdgcn_wmma_f32_16x16x4_f32(false, a, false, b1, (short)0, c1, false, false);
        c2 = __builtin_amdgcn_wmma_f32_16x16x4_f32(false, a, false, b2, (short)0, c2, false, false);
        c3 = __builtin_amdgcn_wmma_f32_16x16x4_f32(false, a, false, b3, (short)0, c3, false, false);
    }

    // C layout: VGPR j -> (M = j + 8*hi, N = lane%16). Dump logits to LDS.
#pragma unroll
    for (int j = 0; j < 8; ++j) {
        const int m = j + 8 * hi;
        sLogits[wv][m][row +  0] = c0[j];
        sLogits[wv][m][row + 16] = c1[j];
        sLogits[wv][m][row + 32] = c2[j];
        sLogits[wv][m][row + 48] = c3[j];
    }
    __syncthreads();

    // --- per-token softmax + grouped top-k (one lane per token) ---
    if (lane < 16) {
        float* srow = sLogits[wv][lane];
        float mx = srow[0];
#pragma unroll
        for (int e = 1; e < NE; ++e) mx = fmaxf(mx, srow[e]);
        float sum = 0.0f;
#pragma unroll
        for (int e = 0; e < NE; ++e) {
            float t = __expf(srow[e] - mx);
            srow[e] = t;                 // store exp; score = srow[e]*inv
            sum += t;
        }
        const float inv = 1.0f / sum;

        // group scores = top-2 sum of (score + bias); accumulate p into LDS
        float gs[NGRP];
#pragma unroll
        for (int g = 0; g < NGRP; ++g) {
            float m1 = -__builtin_inff(), m2 = -__builtin_inff();
#pragma unroll
            for (int j = 0; j < EPG; ++j) {
                const int e = g * EPG + j;
                const float sc = srow[e] * inv;
                atomicAdd(&sP[e], sc);
                const float sb = sc + sBias[e];
                if (sb > m1)      { m2 = m1; m1 = sb; }
                else if (sb > m2) { m2 = sb; }
            }
            gs[g] = m1 + m2;
        }

        // top-4 groups (first-occurrence wins ties, like jax top_k)
        unsigned keep = 0u;
        for (int t = 0; t < TG; ++t) {
            float best = -__builtin_inff(); int bg = 0;
            for (int g = 0; g < NGRP; ++g)
                if (!((keep >> g) & 1u) && gs[g] > best) { best = gs[g]; bg = g; }
            keep |= 1u << bg;
        }

        // top-6 experts over kept groups; weights from ORIGINAL (un-biased) scores
        unsigned long long chosen = 0ull;
        const int token = t0 + lane;
        for (int j = 0; j < NTOPK; ++j) {
            float best = -__builtin_inff(); int be = 0;
            for (int e = 0; e < NE; ++e) {
                if (!((keep >> (e >> 3)) & 1u)) continue;
                if ((chosen >> e) & 1ull)       continue;
                const float v = srow[e] * inv + sBias[e];
                if (v > best) { best = v; be = e; }
            }
            chosen |= 1ull << be;
            outW[token * NTOPK + j] = srow[be] * inv * ROUTE_SCALE_F;
            outI[token * NTOPK + j] = be;
            atomicAdd(&sCnt[be], 1.0f);
        }
    }
    __syncthreads();

    if (tid < NE) {
        atomicAdd(&gCnt[tid], sCnt[tid]);
        atomicAdd(&gP[tid],   sP[tid]);
    }
}

__global__ void moe_loss_kernel(const float* __restrict__ gCnt,
                                const float* __restrict__ gP,
                                float* __restrict__ outLoss)
{
    __shared__ float part[NE];
    const int e = threadIdx.x;
    const float f = gCnt[e] * ((float)NE / ((float)NTOKENS * (float)NTOPK));
    const float p = gP[e] * (1.0f / (float)NTOKENS);
    part[e] = p * f;
    __syncthreads();
    if (e == 0) {
        float acc = 0.0f;
        for (int i = 0; i < NE; ++i) acc += part[i];
        *outLoss = acc * BAL_F;
    }
}

extern "C" void kernel_launch(void* const* d_in, const int* in_sizes, int n_in,
                              void* d_out, int out_size, void* d_ws, size_t ws_size,
                              hipStream_t stream) {
    const float* x      = (const float*)d_in[0];   // [65536, 2048] fp32
    const float* weight = (const float*)d_in[1];   // [64, 2048]    fp32
    const float* bias   = (const float*)d_in[2];   // [64]          fp32

    float* outF   = (float*)d_out;
    float* outW   = outF;                                       // [65536*6] fp32
    int*   outI   = (int*)(outF + (size_t)NTOKENS * NTOPK);     // [65536*6] int32
    float* outLoss = outF + (size_t)NTOKENS * NTOPK * 2;        // [1] fp32

    float* gCnt = (float*)d_ws;       // [64]
    float* gP   = gCnt + NE;          // [64]

    zero_ws_kernel<<<1, 2 * NE, 0, stream>>>(gCnt);
    moe_gate_wmma_kernel<<<NTOKENS / 128, 256, 0, stream>>>(
        x, weight, bias, outW, outI, gCnt, gP);
    moe_loss_kernel<<<1, NE, 0, stream>>>(gCnt, gP, outLoss);
}